// UniformBottomUpHTMM_55731495633410
// MI455X (gfx1250) — compile-verified
//
#include <hip/hip_runtime.h>

typedef __attribute__((ext_vector_type(2))) float v2f;
typedef __attribute__((ext_vector_type(8))) float v8f;

__device__ inline v8f wmma_f32_16x16x4(v2f a, v2f b, v8f c) {
  // D = A(16x4 f32) x B(4x16 f32) + C(16x16 f32)
  return __builtin_amdgcn_wmma_f32_16x16x4_f32(
      /*neg_a=*/false, a, /*neg_b=*/false, b,
      /*c_mod=*/(short)0, c, /*reuse_a=*/false, /*reuse_b=*/false);
}

// a/b for b > 0: v_rcp_f32 + one Newton-Raphson step (~1 ulp) instead of the
// ~10-instruction IEEE v_div_scale/v_div_fmas/v_div_fixup chain.
__device__ inline float fast_div(float a, float b) {
  float r = __builtin_amdgcn_rcpf(b);
  r = __builtin_fmaf(__builtin_fmaf(-b, r, 1.0f), r, r);
  return a * r;
}

// One wave32 per (tree, generator) pair: grid = T * NGEN blocks of 32 threads.
// Bottom-up pass over a complete binary tree (heap layout, 255 nodes); the
// per-level (E x 16) @ (16 x 16) state-transition GEMM runs on f32 WMMA.
__global__ __launch_bounds__(32)
void htmm_upward_kernel(const float* __restrict__ A,   // (C, C, NGEN)
                        const float* __restrict__ B,   // (C, M, NGEN)
                        const float* __restrict__ Pi,  // (C, NGEN)
                        const int*   __restrict__ x,   // (T * NPT,)
                        float* __restrict__ out)       // (T, NGEN)
{
  constexpr int C = 16, M = 32, NGEN = 16, NPT = 255;

  __shared__ float sBuf0[128 * 16];   // beta ping buffer (<=128 nodes x 16 states)
  __shared__ float sBuf1[64 * 16];    // beta pong buffer (<=64 nodes x 16 states)
  __shared__ float sAt[16 * 16];      // smA^T: sAt[c*16 + p] = softmax_p(A[p,c,g])
  __shared__ float sB[16 * 32];       // smB:   sB[c*32 + m]  = softmax_m(B[c,m,g])
  __shared__ float sPi[16];           // smPi
  __shared__ int   sX[256];           // this tree's observation symbols

  const int g     = blockIdx.x & (NGEN - 1);
  const int tree  = blockIdx.x >> 4;
  const int lane  = threadIdx.x;     // 0..31
  const int h     = lane >> 4;       // half-wave id (0/1)
  const int n     = lane & 15;       // lane within half (= state index / N index)
  const int xbase = tree * NPT;

  // zero-fill beta buffers (padding rows of partial WMMA tiles must read zeros)
  for (int i = lane; i < 128 * 16; i += 32) sBuf0[i] = 0.f;
  for (int i = lane; i < 64 * 16;  i += 32) sBuf1[i] = 0.f;
  // preload this tree's x[] once (coalesced); inner loops then use LDS only
  for (int i = lane; i < NPT; i += 32) sX[i] = x[xbase + i];

  // ---------- softmax(Pi[:, g]) over states (lanes 0..15) ----------
  if (lane < 16) {
    float v  = Pi[lane * NGEN + g];
    float mx = v;
    for (int o = 8; o >= 1; o >>= 1) mx = fmaxf(mx, __shfl_xor(mx, o, 16));
    float e = expf(v - mx);
    float s = e;
    for (int o = 8; o >= 1; o >>= 1) s += __shfl_xor(s, o, 16);
    sPi[lane] = fast_div(e, s);
  }
  // ---------- smA^T: for column c, softmax over p of A[p,c,g] ----------
  for (int it = 0; it < 8; ++it) {
    int   c  = it * 2 + h;                 // each half handles one column c
    int   p  = n;
    float v  = A[(p * C + c) * NGEN + g];
    float mx = v;
    for (int o = 8; o >= 1; o >>= 1) mx = fmaxf(mx, __shfl_xor(mx, o, 16));
    float e = expf(v - mx);
    float s = e;
    for (int o = 8; o >= 1; o >>= 1) s += __shfl_xor(s, o, 16);
    sAt[c * 16 + p] = fast_div(e, s);
  }
  // ---------- smB: for each state c, softmax over the 32 symbols ----------
  for (int c = 0; c < C; ++c) {
    float v  = B[(c * M + lane) * NGEN + g];
    float mx = v;
    for (int o = 16; o >= 1; o >>= 1) mx = fmaxf(mx, __shfl_xor(mx, o, 32));
    float e = expf(v - mx);
    float s = e;
    for (int o = 16; o >= 1; o >>= 1) s += __shfl_xor(s, o, 32);
    sB[c * M + lane] = fast_div(e, s);
  }
  __syncthreads();

  // Preload the 4 constant B-operand fragments (smA^T), reused by every tile.
  // B[k=c_child][n=c_parent] = smA[c_parent, c_child] = sAt[c*16 + n].
  v2f bfrag[4];
#pragma unroll
  for (int kk = 0; kk < 4; ++kk) {
    int k0 = kk * 4 + 2 * h;               // K pairs: low half {0,1}, high {2,3}
    bfrag[kk].x = sAt[(k0    ) * 16 + n];
    bfrag[kk].y = sAt[(k0 + 1) * 16 + n];
  }

  float ll  = 0.f;                          // per-lane log-likelihood partial
  float pin = sPi[n];

  // ---------- leaves (level 7): 128 nodes, one per half-wave per pass ----------
  for (int it = 0; it < 64; ++it) {
    int   j   = it * 2 + h;                 // leaf index within level
    int   xv  = sX[127 + j];
    float val = pin * sB[n * M + xv];
    float nu  = val;
    for (int o = 8; o >= 1; o >>= 1) nu += __shfl_xor(nu, o, 16);
    sBuf0[j * 16 + n] = fast_div(val, nu);
    if (n == 0) ll += logf(nu);
  }
  __syncthreads();

  float* cur = sBuf0;                       // child-level betas
  float* nxt = sBuf1;                       // parent-level betas

  // ---------- internal levels l = 6 .. 0 ----------
  for (int l = 6; l >= 0; --l) {
    int P     = 1 << l;                     // parents at level l
    int E     = P << 1;                     // children at level l+1
    int tiles = (E + 15) >> 4;
    for (int t = 0; t < tiles; ++t) {
      v8f acc = {0.f, 0.f, 0.f, 0.f, 0.f, 0.f, 0.f, 0.f};
      // A[m=child_row][k=c_child] = cur[row*16 + c]; f32 16x4 layout:
      // lanes 0-15 hold K pair {k0, k0+1}, lanes 16-31 hold {k0+2, k0+3}.
#pragma unroll
      for (int kk = 0; kk < 4; ++kk) {
        int row = t * 16 + n;
        int k0  = kk * 4 + 2 * h;
        v2f a;
        a.x = cur[row * 16 + k0];
        a.y = cur[row * 16 + k0 + 1];
        acc = wmma_f32_16x16x4(a, bfrag[kk], acc);
      }
      // D rows r and r+1 (sibling children) live in acc[r], acc[r+1] of one
      // lane; low half covers rows 0-7 (parents t*8+0..3), high half rows
      // 8-15 (parents t*8+4..7). Average, apply emission, normalize, log.
#pragma unroll
      for (int j = 0; j < 4; ++j) {
        int pl = t * 8 + h * 4 + j;         // parent index within level l
        if (pl < P) {                        // uniform per half-wave
          float tv = 0.5f * (acc[2 * j] + acc[2 * j + 1]);
          int   xv = sX[(P - 1) + pl];       // LVL_START[l] = 2^l - 1
          float bb = sB[n * M + xv] * tv;
          float nu = bb;
          for (int o = 8; o >= 1; o >>= 1) nu += __shfl_xor(nu, o, 16);
          nxt[pl * 16 + n] = fast_div(bb, nu);
          if (n == 0) ll += logf(nu);
        }
      }
    }
    __syncthreads();
    float* tmp = cur; cur = nxt; nxt = tmp;
  }

  // ---------- reduce ll across the wave, emit out[tree, g] ----------
  for (int o = 16; o >= 1; o >>= 1) ll += __shfl_xor(ll, o, 32);
  if (lane == 0) out[tree * NGEN + g] = ll;
}

extern "C" void kernel_launch(void* const* d_in, const int* in_sizes, int n_in,
                              void* d_out, int out_size, void* d_ws, size_t ws_size,
                              hipStream_t stream) {
  (void)n_in; (void)d_ws; (void)ws_size; (void)in_sizes;
  const float* A  = (const float*)d_in[0];   // (16,16,16)
  const float* B  = (const float*)d_in[1];   // (16,32,16)
  const float* Pi = (const float*)d_in[2];   // (16,16)
  const int*   x  = (const int*)d_in[3];     // (T*255,)
  float* out = (float*)d_out;                // (T,16)

  const int T = out_size / 16;               // 2000 trees
  htmm_upward_kernel<<<T * 16, 32, 0, stream>>>(A, B, Pi, x, out);
}